// LogPolar_8091718385906
// MI455X (gfx1250) — compile-verified
//
#include <hip/hip_runtime.h>
#include <math.h>

// Log-polar resample: out[b,c,theta,r] = 4-tap inverse-map gather of data[b,c,:,:]
// Map (X,Y,weights,indices) depends only on (theta,r): compute once per thread,
// reuse across all B*C = 96 planes.

#define H_IN   512
#define W_IN   512
#define H_OUT  512
#define W_OUT  512
#define PLANES 96                      // BATCH(32) * CHANNELS(3)
#define PLANE_STRIDE (H_IN * W_IN)     // 262144 elements, 1 MB per plane

__global__ __launch_bounds__(256) void logpolar_gather_kernel(
    const float* __restrict__ data,   // [96, 512, 512]
    float* __restrict__ out)          // [96, 512, 512]
{
    const int pix = blockIdx.x * blockDim.x + threadIdx.x;  // 0 .. 262143
    const int i = pix >> 9;      // theta index (row)
    const int j = pix & 511;     // r index (col) -> consecutive lanes, coalesced stores

    // ---- per-pixel map, via native CDNA5 transcendentals ----
    // MAX_R = log(||(512,512)||/2 * 2.0) = log(2^9.5) = 9.5*ln2, so
    //   radius = exp(j*MAX_R/512) = 2^(j * 9.5/512)        (exact scale: 0.0185546875)
    // v_exp_f32 computes 2^x; v_sin/cos_f32 take revolutions (sin(2*pi*x)):
    //   sin(i*2*pi/512) = v_sin_f32(i/512)
    const float radius = __builtin_amdgcn_exp2f((float)j * (9.5f / 512.0f));
    const float rev    = (float)i * (1.0f / 512.0f);        // in [0,1): ideal HW range
    const float s      = __builtin_amdgcn_sinf(rev);
    const float c      = __builtin_amdgcn_cosf(rev);

    const float X = 256.0f + radius * c;   // center_x + X0
    const float Y = 256.0f - radius * s;   // center_y - Y0

    const float mask =
        (X >= 0.0f && X < (float)H_IN && Y >= 0.0f && Y < (float)W_IN) ? 1.0f : 0.0f;

    // trunc-toward-zero (C int cast), then clamp — matches reference .astype(int32)+clip
    int y0 = (int)Y; y0 = y0 < 0 ? 0 : (y0 > H_IN - 1 ? H_IN - 1 : y0);
    int x0 = (int)X; x0 = x0 < 0 ? 0 : (x0 > W_IN - 1 ? W_IN - 1 : x0);
    const int y1 = (y0 + 1 > H_IN - 1) ? H_IN - 1 : y0 + 1;
    const int x1 = (x0 + 1 > W_IN - 1) ? W_IN - 1 : x0 + 1;

    const float yd = Y - (float)y0, yu = Y - (float)y1;
    const float xd = X - (float)x0, xu = X - (float)x1;

    const float dd = yd * yd + xd * xd;   // weight for (y0,x0)
    const float du = yd * yd + xu * xu;   // weight for (y0,x1)
    const float ud = yu * yu + xd * xd;   // weight for (y1,x0)
    const float uu = yu * yu + xu * xu;   // weight for (y1,x1)

    const float inv = mask / (dd + du + ud + uu);  // fold mask into the weights
    const float w00 = dd * inv;
    const float w01 = du * inv;
    const float w10 = ud * inv;
    const float w11 = uu * inv;

    const int o00 = y0 * W_IN + x0;
    const int o01 = y0 * W_IN + x1;
    const int o10 = y1 * W_IN + x0;
    const int o11 = y1 * W_IN + x1;

    // ---- plane loop: 4 gathers (L2-resident input) + 1 NT store per plane ----
    const float* __restrict__ p = data;
    float* __restrict__ q = out + pix;

    // Main body: branch-free, prefetch 2 planes ahead (gfx1250 global_prefetch_b8).
    // o00/o01 share a cacheline almost always, as do o10/o11, so 2 prefetches cover
    // both gather rows.
#pragma unroll 4
    for (int pl = 0; pl < PLANES - 2; ++pl) {
        const float* pn = p + 2 * PLANE_STRIDE;
        __builtin_prefetch(pn + o00, 0, 3);
        __builtin_prefetch(pn + o10, 0, 3);

        const float v00 = p[o00];
        const float v01 = p[o01];
        const float v10 = p[o10];
        const float v11 = p[o11];

        const float r = fmaf(w00, v00,
                        fmaf(w01, v01,
                        fmaf(w10, v10, w11 * v11)));

        // Non-temporal store: keep the 100 MB input resident in the 192 MB L2
        // instead of letting the write stream evict it.
        __builtin_nontemporal_store(r, q);

        p += PLANE_STRIDE;
        q += PLANE_STRIDE;
    }

    // Tail: last 2 planes, no prefetch.
#pragma unroll
    for (int pl = 0; pl < 2; ++pl) {
        const float v00 = p[o00];
        const float v01 = p[o01];
        const float v10 = p[o10];
        const float v11 = p[o11];

        const float r = fmaf(w00, v00,
                        fmaf(w01, v01,
                        fmaf(w10, v10, w11 * v11)));

        __builtin_nontemporal_store(r, q);

        p += PLANE_STRIDE;
        q += PLANE_STRIDE;
    }
}

extern "C" void kernel_launch(void* const* d_in, const int* in_sizes, int n_in,
                              void* d_out, int out_size, void* d_ws, size_t ws_size,
                              hipStream_t stream) {
    (void)in_sizes; (void)n_in; (void)out_size; (void)d_ws; (void)ws_size;
    const float* data = (const float*)d_in[0];   // [32,3,512,512] float32
    float* out = (float*)d_out;                  // [32,3,512,512] float32

    const int threads = 256;                               // 8 wave32s per block
    const int blocks  = (H_OUT * W_OUT) / threads;         // 1024 blocks
    logpolar_gather_kernel<<<blocks, threads, 0, stream>>>(data, out);
}